// Encoder_L96_9010841387202
// MI455X (gfx1250) — compile-verified
//
#include <hip/hip_runtime.h>
#include <cstdint>

#define L96_DT 0.01f

typedef unsigned int u32;
typedef unsigned long long u64;
typedef __attribute__((ext_vector_type(4))) u32 v4u;
typedef __attribute__((ext_vector_type(8))) u32 v8u;

// One block per batch element b. Tile = 40 (N) x 64 (T) f32 = 10240 B, contiguous.
__global__ __launch_bounds__(256) void l96_rk4_kernel(const float* __restrict__ x,
                                                      const float* __restrict__ Fp,
                                                      float* __restrict__ out) {
    __shared__ float xs[2560];   // input tile (persistent through all stages)
    __shared__ float vs[2560];   // current RK4 stage state

    const int tid = threadIdx.x;
    const u64 b = blockIdx.x;

    // ---- TDM: DMA this block's contiguous 10240-byte tile into LDS ----
    if (tid < 32) {  // wave 0 only (uniform branch); TDM ignores EXEC anyway
        u64 gaddr = (u64)(uintptr_t)x + b * (2560ull * 4ull);
        u32 lds_off = (u32)(uintptr_t)(&xs[0]);  // low 32 bits of LDS-aperture addr = LDS byte offset

        v4u g0;
        g0.x = 1u;                                   // count=1, user descriptor
        g0.y = lds_off;                              // lds_addr[31:0]
        g0.z = (u32)(gaddr & 0xFFFFFFFFull);         // global_addr[31:0]
        g0.w = ((u32)(gaddr >> 32) & 0x01FFFFFFu)    // global_addr[56:32]
               | (2u << 30);                         // type=2 ("image")

        v8u g1;
        g1[0] = (2u << 16);       // workgroup_mask=0, data_size=2 (4 bytes)
        g1[1] = (2560u << 16);    // tensor_dim0[15:0] in bits 63:48 (=2560)
        g1[2] = (1u << 16);       // tensor_dim0 hi=0 ; tensor_dim1=1 (bits 95:80)
        g1[3] = (2560u << 16);    // tensor_dim1 hi=0 ; tile_dim0=2560 (bits 127:112)
        g1[4] = 1u;               // tile_dim1=1, tile_dim2=0
        g1[5] = 2560u;            // tensor_dim0_stride lo
        g1[6] = 0u;               // tensor_dim0_stride hi, tensor_dim1_stride lo
        g1[7] = 0u;               // tensor_dim1_stride hi

        asm volatile("tensor_load_to_lds %0, %1" :: "s"(g0), "s"(g1) : "memory");
        __builtin_amdgcn_s_wait_tensorcnt(0);
    }
    __syncthreads();

    const float f = Fp[0];
    const int t  = tid & 63;
    const int n0 = tid >> 6;   // 0..3 ; each thread owns n = n0 + 4k, k=0..9

    // Precompute LDS word offsets (own + 3 circular neighbors along N)
    int eo[10], op1[10], om1[10], om2[10];
#pragma unroll
    for (int k = 0; k < 10; ++k) {
        int n   = n0 + 4 * k;
        int np1 = (n == 39) ? 0      : n + 1;
        int nm1 = (n == 0 ) ? 39     : n - 1;
        int nm2 = (n <= 1 ) ? n + 38 : n - 2;
        eo[k]  = n   * 64 + t;
        op1[k] = np1 * 64 + t;
        om1[k] = nm1 * 64 + t;
        om2[k] = nm2 * 64 + t;
    }

    float x_r[10], acc[10], k_r[10];

    // ---- k1 = rhs(x) ; v1 = x + 0.5*DT*k1 ----
#pragma unroll
    for (int k = 0; k < 10; ++k) {
        float v  = xs[eo[k]];
        x_r[k]   = v;
        float kk = (xs[op1[k]] - xs[om2[k]]) * xs[om1[k]] - v + f;
        acc[k]   = kk;
        k_r[k]   = kk;
    }
#pragma unroll
    for (int k = 0; k < 10; ++k) vs[eo[k]] = x_r[k] + (0.5f * L96_DT) * k_r[k];
    __syncthreads();

    // ---- k2 = rhs(v1) ; v2 = x + 0.5*DT*k2 ----
#pragma unroll
    for (int k = 0; k < 10; ++k)
        k_r[k] = (vs[op1[k]] - vs[om2[k]]) * vs[om1[k]] - vs[eo[k]] + f;
    __syncthreads();   // all reads of v1 complete
#pragma unroll
    for (int k = 0; k < 10; ++k) {
        acc[k] += 2.0f * k_r[k];
        vs[eo[k]] = x_r[k] + (0.5f * L96_DT) * k_r[k];
    }
    __syncthreads();   // v2 visible

    // ---- k3 = rhs(v2) ; v3 = x + DT*k3 ----
#pragma unroll
    for (int k = 0; k < 10; ++k)
        k_r[k] = (vs[op1[k]] - vs[om2[k]]) * vs[om1[k]] - vs[eo[k]] + f;
    __syncthreads();
#pragma unroll
    for (int k = 0; k < 10; ++k) {
        acc[k] += 2.0f * k_r[k];
        vs[eo[k]] = x_r[k] + L96_DT * k_r[k];
    }
    __syncthreads();   // v3 visible

    // ---- k4 = rhs(v3) ; combine + store ----
#pragma unroll
    for (int k = 0; k < 10; ++k)
        k_r[k] = (vs[op1[k]] - vs[om2[k]]) * vs[om1[k]] - vs[eo[k]] + f;

    float* ob = out + b * 2600ull;   // 40 * 65 per batch
#pragma unroll
    for (int k = 0; k < 10; ++k) {
        acc[k] += k_r[k];
        float xp = x_r[k] + (L96_DT * (1.0f / 6.0f)) * acc[k];
        int n = n0 + 4 * k;
        ob[n * 65 + t + 1] = xp;          // xpred goes to time slots 1..64
        if (t == 0) ob[n * 65] = x_r[k];  // prepended first input slice
    }
}

extern "C" void kernel_launch(void* const* d_in, const int* in_sizes, int n_in,
                              void* d_out, int out_size, void* d_ws, size_t ws_size,
                              hipStream_t stream) {
    (void)n_in; (void)out_size; (void)d_ws; (void)ws_size;
    const float* x = (const float*)d_in[0];
    const float* F = (const float*)d_in[1];
    float* out     = (float*)d_out;
    const int B = in_sizes[0] / 2560;   // 8192 for the reference shapes
    l96_rk4_kernel<<<dim3(B), dim3(256), 0, stream>>>(x, F, out);
}